// MeshUnpool_34299608826682
// MI455X (gfx1250) — compile-verified
//
#include <hip/hip_runtime.h>
#include <hip/hip_bf16.h>
#include <limits.h>

// ---------------------------------------------------------------------------
// MeshUnpool for MI455X (gfx1250, wave32).
// Zero-FLOP op: resolve K sequential row-copies into a per-vertex source
// pointer with O(N) parallel pointer chasing (int32 metadata, ~5MB traffic),
// then stream the 256MB row gather through the CDNA5 async Global<->LDS
// engine (global_load_async_to_lds_b128 / global_store_async_from_lds_b128,
// ASYNCcnt) at HBM rate.
// ---------------------------------------------------------------------------

#define THREADS 256          // 8 wave32 waves per block
#define SCAN_ELEMS 1024      // elements scanned per block (4 per thread)

__global__ void k_zero(int* p, int n) {
  int i = blockIdx.x * blockDim.x + threadIdx.x;
  if (i < n) p[i] = 0;
}

__global__ void k_copy(int* dst, const int* src, int n) {
  int i = blockIdx.x * blockDim.x + threadIdx.x;
  if (i < n) dst[i] = src[i];
}

__global__ void k_mask_to_int(const unsigned char* mask, int* mi, int n) {
  int i = blockIdx.x * blockDim.x + threadIdx.x;
  if (i < n) mi[i] = mask[i] ? 1 : 0;
}

// Histogram of write targets: row 1 of order is t (destination vertex).
__global__ void k_count(const int* order, int* counts, int K, int N) {
  int j = blockIdx.x * blockDim.x + threadIdx.x;
  if (j < K) {
    int t = order[(size_t)K + j];
    if ((unsigned)t < (unsigned)N) atomicAdd(&counts[t], 1);
  }
}

// In-place exclusive scan of a 1024-element chunk per block; emits chunk total.
__global__ void k_scan_block(int* data, int* blockSums, int n) {
  __shared__ int sdata[THREADS];
  const int t = threadIdx.x;
  const int base = blockIdx.x * SCAN_ELEMS + t * 4;
  int v0 = (base + 0 < n) ? data[base + 0] : 0;
  int v1 = (base + 1 < n) ? data[base + 1] : 0;
  int v2 = (base + 2 < n) ? data[base + 2] : 0;
  int v3 = (base + 3 < n) ? data[base + 3] : 0;
  int s0 = v0, s1 = s0 + v1, s2 = s1 + v2, s3 = s2 + v3;
  sdata[t] = s3;
  __syncthreads();
  for (int off = 1; off < THREADS; off <<= 1) {
    int x = (t >= off) ? sdata[t - off] : 0;
    __syncthreads();
    sdata[t] += x;
    __syncthreads();
  }
  int excl = (t > 0) ? sdata[t - 1] : 0;
  if (base + 0 < n) data[base + 0] = excl;
  if (base + 1 < n) data[base + 1] = excl + s0;
  if (base + 2 < n) data[base + 2] = excl + s1;
  if (base + 3 < n) data[base + 3] = excl + s2;
  if (t == THREADS - 1) blockSums[blockIdx.x] = sdata[THREADS - 1];
}

// Single-block exclusive scan over the per-block totals.
__global__ void k_scan_sums(int* bs, int nBlocks) {
  __shared__ int sdata[THREADS];
  int carry = 0;
  for (int start = 0; start < nBlocks; start += THREADS) {
    int i = start + threadIdx.x;
    int v = (i < nBlocks) ? bs[i] : 0;
    sdata[threadIdx.x] = v;
    __syncthreads();
    for (int off = 1; off < THREADS; off <<= 1) {
      int x = (threadIdx.x >= off) ? sdata[threadIdx.x - off] : 0;
      __syncthreads();
      sdata[threadIdx.x] += x;
      __syncthreads();
    }
    int excl = ((threadIdx.x > 0) ? sdata[threadIdx.x - 1] : 0) + carry;
    int total = sdata[THREADS - 1];
    __syncthreads();
    if (i < nBlocks) bs[i] = excl;
    carry += total;
  }
}

__global__ void k_scan_add(int* data, const int* blockSums, int n) {
  int i = blockIdx.x * blockDim.x + threadIdx.x;
  if (i < n) data[i] += blockSums[i / SCAN_ELEMS];
}

// Scatter column indices into CSR buckets (order within a bucket is
// nondeterministic; the resolver takes the min, so results are deterministic).
__global__ void k_fill(const int* order, int* cursor, int* colIdx, int K, int N) {
  int j = blockIdx.x * blockDim.x + threadIdx.x;
  if (j < K) {
    int t = order[(size_t)K + j];
    if ((unsigned)t < (unsigned)N) {
      int slot = atomicAdd(&cursor[t], 1);
      colIdx[slot] = j;
    }
  }
}

// Resolve final source vertex for each output row.
// Columns are processed j = K-1 ... 0; src[t_j] = src[f_j].
// resolve(v): j* = min { j in bucket(v) : j >= bound }; none -> v;
//             else hop to (f_{j*}, bound = j*+1).
__global__ void k_resolve(const int* order, const int* offsets, const int* ends,
                          const int* colIdx, int* srcArr, int N, int K) {
  int v = blockIdx.x * blockDim.x + threadIdx.x;
  if (v >= N) return;
  int cur = v;
  int bound = 0;
  for (int iter = 0; iter <= K; ++iter) {
    int s = offsets[cur];
    int e = ends[cur];
    int best = INT_MAX;
    for (int p = s; p < e; ++p) {
      int j = colIdx[p];
      if (j >= bound && j < best) best = j;
    }
    if (best == INT_MAX) break;
    int f = order[best];            // row 0 of order
    bound = best + 1;
    if ((unsigned)f >= (unsigned)N) break;
    cur = f;
  }
  srcArr[v] = cur;
}

// Collapse mask/pos/src into one per-row img-row index (-1 => zero row).
__global__ void k_rowsrc(const unsigned char* mask, const int* pos,
                         const int* srcArr, int* rsrc, int N, int n_active) {
  int v = blockIdx.x * blockDim.x + threadIdx.x;
  if (v >= N) return;
  int s = srcArr[v];
  int r = -1;
  if (mask[s]) {
    int p = pos[s];
    if (p < 0) p = 0;
    if (p >= n_active) p = n_active - 1;
    r = p;
  }
  rsrc[v] = r;
}

// ---------------------------------------------------------------------------
// CDNA5 async-LDS row gather (C == 256 floats => 1KB rows).
// One wave per row iteration: two global_load_async_to_lds_b128 (32 lanes x
// 16B x 2 = 1KB) into a per-wave LDS buffer, s_wait_asynccnt, then two
// global_store_async_from_lds_b128 to the output row. Zero rows write zero
// float4s directly (no LDS). The row index is wave-uniform, so the branch is
// wave-uniform and EXEC stays all-ones for the async ops.
// ---------------------------------------------------------------------------
__global__ void k_gather_async(const float* __restrict__ img,
                               const int* __restrict__ rsrc,
                               float* __restrict__ out, int N) {
  __shared__ __align__(16) unsigned char ldsbuf[8 * 1024];  // 1KB per wave
  const int lane = threadIdx.x & 31;
  const int waveInBlk = threadIdx.x >> 5;
  const int wavesPerBlk = blockDim.x >> 5;
  const int wid = blockIdx.x * wavesPerBlk + waveInBlk;
  const int nWaves = gridDim.x * wavesPerBlk;

  // Flat shared pointers truncate to the workgroup-relative LDS byte offset.
  const unsigned ldsBase = (unsigned)(size_t)(&ldsbuf[waveInBlk * 1024]);
  const unsigned lds0 = ldsBase + (unsigned)(lane * 16);
  const unsigned lds1 = lds0 + 512u;
  const unsigned long long gin = (unsigned long long)(size_t)img;
  const unsigned long long gout = (unsigned long long)(size_t)out;

  for (int r = wid; r < N; r += nWaves) {
    int p = rsrc[r];
    if (p >= 0) {
      int vin0 = p * 1024 + lane * 16;
      int vin1 = vin0 + 512;
      int vo0 = r * 1024 + lane * 16;
      int vo1 = vo0 + 512;
      // Drain prior async stores before overwriting this wave's LDS buffer.
      asm volatile("s_wait_asynccnt 0" ::: "memory");
      asm volatile("global_load_async_to_lds_b128 %0, %1, %2"
                   :: "v"(lds0), "v"(vin0), "s"(gin) : "memory");
      asm volatile("global_load_async_to_lds_b128 %0, %1, %2"
                   :: "v"(lds1), "v"(vin1), "s"(gin) : "memory");
      asm volatile("s_wait_asynccnt 0" ::: "memory");
      asm volatile("global_store_async_from_lds_b128 %0, %1, %2"
                   :: "v"(vo0), "v"(lds0), "s"(gout) : "memory");
      asm volatile("global_store_async_from_lds_b128 %0, %1, %2"
                   :: "v"(vo1), "v"(lds1), "s"(gout) : "memory");
    } else {
      float4 z = make_float4(0.f, 0.f, 0.f, 0.f);
      float4* o = (float4*)(out + (size_t)r * 256);
      o[lane] = z;
      o[lane + 32] = z;
    }
  }
  // Drain outstanding async stores before the wave exits.
  asm volatile("s_wait_asynccnt 0" ::: "memory");
}

// Generic fallback gather (any C % 4 == 0): one float4 per thread.
__global__ void k_gather(const float* __restrict__ img,
                         const int* __restrict__ rsrc,
                         float* __restrict__ out, int N, int vec) {
  long long g = (long long)blockIdx.x * blockDim.x + threadIdx.x;
  long long total = (long long)N * vec;
  if (g >= total) return;
  int row = (int)(g / vec);
  int lane = (int)(g % vec);
  int p = rsrc[row];
  float4 val = make_float4(0.f, 0.f, 0.f, 0.f);
  if (p >= 0) {
    val = ((const float4*)img)[(long long)p * vec + lane];
  }
  ((float4*)out)[g] = val;
}

extern "C" void kernel_launch(void* const* d_in, const int* in_sizes, int n_in,
                              void* d_out, int out_size, void* d_ws, size_t ws_size,
                              hipStream_t stream) {
  const float* img = (const float*)d_in[0];
  const unsigned char* mask = (const unsigned char*)d_in[1];  // jnp.bool_ = 1 byte
  const int* order = (const int*)d_in[2];

  const int N = in_sizes[1];            // 262144
  const int K = in_sizes[2] / 2;        // 131072
  const int C = out_size / N;           // 256
  const int vec = C / 4;                // float4 lanes per row
  const int n_active = in_sizes[0] / C; // 131072

  float* out = (float*)d_out;

  // Workspace layout (ints): counts/offsets | cursor/ends | colIdx | src | maskInt/pos | blockSums
  int* W = (int*)d_ws;
  int* counts  = W;                       // N   -> exclusive-scanned to offsets; reused as rsrc
  int* cursor  = W + (size_t)N;           // N   -> bucket ends after fill
  int* colIdx  = W + (size_t)2 * N;       // K
  int* srcArr  = W + (size_t)2 * N + K;   // N
  int* maskInt = W + (size_t)3 * N + K;   // N   -> exclusive-scanned to pos
  int* bsums   = W + (size_t)4 * N + K;   // up to 1024

  const int gN = (N + THREADS - 1) / THREADS;
  const int gK = (K + THREADS - 1) / THREADS;
  const int nChunks = (N + SCAN_ELEMS - 1) / SCAN_ELEMS;   // 256

  // 1) Histogram of write targets.
  k_zero<<<gN, THREADS, 0, stream>>>(counts, N);
  k_count<<<gK, THREADS, 0, stream>>>(order, counts, K, N);

  // 2) Exclusive scan -> CSR offsets.
  k_scan_block<<<nChunks, THREADS, 0, stream>>>(counts, bsums, N);
  k_scan_sums<<<1, THREADS, 0, stream>>>(bsums, nChunks);
  k_scan_add<<<gN, THREADS, 0, stream>>>(counts, bsums, N);

  // 3) Fill buckets; cursor ends up as bucket end pointers.
  k_copy<<<gN, THREADS, 0, stream>>>(cursor, counts, N);
  k_fill<<<gK, THREADS, 0, stream>>>(order, cursor, colIdx, K, N);

  // 4) pos = exclusive prefix sum of mask.
  k_mask_to_int<<<gN, THREADS, 0, stream>>>(mask, maskInt, N);
  k_scan_block<<<nChunks, THREADS, 0, stream>>>(maskInt, bsums, N);
  k_scan_sums<<<1, THREADS, 0, stream>>>(bsums, nChunks);
  k_scan_add<<<gN, THREADS, 0, stream>>>(maskInt, bsums, N);

  // 5) Resolve the copy-chain pointer for every output row.
  k_resolve<<<gN, THREADS, 0, stream>>>(order, counts, cursor, colIdx, srcArr, N, K);

  // 6) Per-row source index (reuse counts buffer; offsets no longer needed).
  k_rowsrc<<<gN, THREADS, 0, stream>>>(mask, maskInt, srcArr, counts, N, n_active);

  // 7) Bandwidth-bound row gather (~16us floor; everything above is noise).
  if (C == 256) {
    // CDNA5 async Global<->LDS streaming path (1KB rows).
    k_gather_async<<<2048, THREADS, 0, stream>>>(img, counts, out, N);
  } else {
    long long totalVec = (long long)N * vec;
    int gG = (int)((totalVec + THREADS - 1) / THREADS);
    k_gather<<<gG, THREADS, 0, stream>>>(img, counts, out, N, vec);
  }
}